// PDHGSolver_35347580846319
// MI455X (gfx1250) — compile-verified
//
#include <hip/hip_runtime.h>

// PDHG solver: B=8192, N=256, M1=128, M2=64, 30 iterations, fp32 WMMA 16x16x4.
// One block = 4 waves = two 16-row tiles; A1||A2 (192x256) resident in LDS.

typedef __attribute__((ext_vector_type(2))) float v2f;
typedef __attribute__((ext_vector_type(8))) float v8f;

#define TAU   0.05f
#define SIGMA 0.05f
#define MAX_ITER 30

static constexpr int AP = 260;  // pitch (dwords) for A, x, xbar  (256 + 4 pad)
static constexpr int YP = 196;  // pitch (dwords) for ys, z       (192 + 4 pad)

static constexpr int OFF_A  = 0;                   // [192][AP]
static constexpr int OFF_X  = OFF_A  + 192 * AP;   // [2][16][AP]
static constexpr int OFF_XB = OFF_X  + 32  * AP;   // [2][16][AP]
static constexpr int OFF_YS = OFF_XB + 32  * AP;   // [2][16][YP]  holds -TAU*y
static constexpr int OFF_Z  = OFF_YS + 32  * YP;   // [2][16][YP]
static constexpr int OFF_C  = OFF_Z  + 32  * YP;   // [256]
static constexpr int SMEM_TOT = OFF_C + 256;       // 79360 dwords = 317440 B <= 320KB

__global__ __launch_bounds__(128, 1)
void pdhg_kernel(const float* __restrict__ x0g, const float* __restrict__ y1g,
                 const float* __restrict__ y2g, const float* __restrict__ zg,
                 const float* __restrict__ cg,  const float* __restrict__ A1,
                 const float* __restrict__ A2,  float* __restrict__ out)
{
    __shared__ float smem[SMEM_TOT];

    const int tid  = threadIdx.x;
    const int wave = tid >> 5;
    const int lane = tid & 31;
    const int tile = wave >> 1;     // 0/1: which 16-row tile
    const int half = wave & 1;      // 0/1: which column half of the tile
    const int n    = lane & 15;     // output column within 16x16 tile
    const int hi   = (lane >> 4) << 3;   // row offset for C-layout (0 or 8)
    const int ko2  = (lane >> 4) << 1;   // K offset for A/B frags (0 or 2)
    const int pid  = (half << 5) | lane; // 0..63 within the wave-pair

    const int rbase = blockIdx.x * 32 + tile * 16;
    const int xw  = OFF_X  + tile * 16 * AP;
    const int xbw = OFF_XB + tile * 16 * AP;
    const int ysw = OFF_YS + tile * 16 * YP;
    const int zw  = OFF_Z  + tile * 16 * YP;

    // ---------------- cooperative loads ----------------
    // A1||A2 -> padded [192][AP], all 128 threads
    for (int i = tid; i < 192 * 64; i += 128) {
        int r = i >> 6, c4 = (i & 63) << 2;
        const float* src = (r < 128) ? (A1 + r * 256 + c4) : (A2 + (r - 128) * 256 + c4);
        *(float4*)&smem[OFF_A + r * AP + c4] = *(const float4*)src;
    }
    // c (uniform per-wave branch)
    if (tid < 64)
        *(float4*)&smem[OFF_C + tid * 4] = *(const float4*)(cg + tid * 4);
    // x0 -> x and xbar (wave-pair cooperates on its tile)
    for (int i = pid; i < 16 * 64; i += 64) {
        int r = i >> 6, c4 = (i & 63) << 2;
        float4 v = *(const float4*)(x0g + (size_t)(rbase + r) * 256 + c4);
        *(float4*)&smem[xw  + r * AP + c4] = v;
        *(float4*)&smem[xbw + r * AP + c4] = v;
    }
    // z tile (float2 copies, 96 per row)
    for (int i = pid; i < 16 * 96; i += 64) {
        int r = i / 96, c2 = (i % 96) << 1;
        float2 v = *(const float2*)(zg + (size_t)(rbase + r) * 192 + c2);
        *(float2*)&smem[zw + r * YP + c2] = v;
    }
    // y0 -> ys scaled by -TAU (y1 cols 0..127, y2 cols 128..191)
    for (int i = pid; i < 16 * 96; i += 64) {
        int r = i / 96, c2 = (i % 96) << 1;
        float2 v;
        if (c2 < 128) v = *(const float2*)(y1g + (size_t)(rbase + r) * 128 + c2);
        else          v = *(const float2*)(y2g + (size_t)(rbase + r) * 64 + (c2 - 128));
        v.x *= -TAU; v.y *= -TAU;
        *(float2*)&smem[ysw + r * YP + c2] = v;
    }
    __syncthreads();

    const float NEG_INV_TS = -1.0f / (TAU * SIGMA);   // ys -> y/SIGMA

    for (int it = 0; it < MAX_ITER; ++it) {
        // ===== Phase A: acc = y/SIGMA - z + xbar @ Acat^T  (this wave: 6 col tiles) =====
        v8f acc[6];
#pragma unroll
        for (int cti = 0; cti < 6; ++cti) {
            int ct = half * 6 + cti;
#pragma unroll
            for (int v = 0; v < 8; ++v) {
                int row = v + hi, col = ct * 16 + n;
                acc[cti][v] = smem[ysw + row * YP + col] * NEG_INV_TS
                            - smem[zw  + row * YP + col];
            }
        }
        for (int k0 = 0; k0 < 256; k0 += 4) {
            int ko = k0 + ko2;
            v2f a = *(const v2f*)&smem[xbw + (lane & 15) * AP + ko];
#pragma unroll
            for (int cti = 0; cti < 6; ++cti) {
                int ct = half * 6 + cti;
                v2f b = *(const v2f*)&smem[OFF_A + (ct * 16 + (lane & 15)) * AP + ko];
                acc[cti] = __builtin_amdgcn_wmma_f32_16x16x4_f32(
                    false, a, false, b, (short)0, acc[cti], false, false);
            }
        }
        // y update: y = SIGMA*acc (relu for cols < 128); store ys = -TAU*y
#pragma unroll
        for (int cti = 0; cti < 6; ++cti) {
            int ct = half * 6 + cti;
#pragma unroll
            for (int v = 0; v < 8; ++v) {
                float yv = SIGMA * acc[cti][v];
                yv = (ct < 8) ? fmaxf(yv, 0.0f) : yv;
                smem[ysw + (v + hi) * YP + ct * 16 + n] = -TAU * yv;
            }
        }
        __syncthreads();

        // ===== Phase B: acc2 = x - TAU*c + ys @ Acat  (this wave: 8 col tiles) =====
        v8f acc2[8];
#pragma unroll
        for (int cti = 0; cti < 8; ++cti) {
            int ct = half * 8 + cti;
            float cv = smem[OFF_C + ct * 16 + n];
#pragma unroll
            for (int v = 0; v < 8; ++v)
                acc2[cti][v] = smem[xw + (v + hi) * AP + ct * 16 + n] - TAU * cv;
        }
        for (int m0 = 0; m0 < 192; m0 += 4) {
            int mo = m0 + ko2;
            v2f a = *(const v2f*)&smem[ysw + (lane & 15) * YP + mo];
#pragma unroll
            for (int cti = 0; cti < 8; ++cti) {
                int ct = half * 8 + cti;
                v2f b;
                b.x = smem[OFF_A + mo * AP + ct * 16 + n];
                b.y = smem[OFF_A + (mo + 1) * AP + ct * 16 + n];
                acc2[cti] = __builtin_amdgcn_wmma_f32_16x16x4_f32(
                    false, a, false, b, (short)0, acc2[cti], false, false);
            }
        }
        // x_next = max(acc2, 0); xbar = 2*x_next - x_old
#pragma unroll
        for (int cti = 0; cti < 8; ++cti) {
            int ct = half * 8 + cti;
#pragma unroll
            for (int v = 0; v < 8; ++v) {
                int idx = (v + hi) * AP + ct * 16 + n;
                float xo = smem[xw + idx];
                float xn = fmaxf(acc2[cti][v], 0.0f);
                smem[xw  + idx] = xn;
                smem[xbw + idx] = 2.0f * xn - xo;
            }
        }
        __syncthreads();
    }

    // ---------------- outputs: concat(x, y1, y2) ----------------
    float* out_x  = out;
    float* out_y1 = out + (size_t)8192 * 256;
    float* out_y2 = out_y1 + (size_t)8192 * 128;
    for (int i = pid; i < 16 * 64; i += 64) {
        int r = i >> 6, c4 = (i & 63) << 2;
        float4 v = *(const float4*)&smem[xw + r * AP + c4];
        *(float4*)(out_x + (size_t)(rbase + r) * 256 + c4) = v;
    }
    const float NEG_INV_TAU = -1.0f / TAU;
    for (int i = pid; i < 16 * 96; i += 64) {
        int r = i / 96, c2 = (i % 96) << 1;
        float2 v = *(const float2*)&smem[ysw + r * YP + c2];
        v.x *= NEG_INV_TAU; v.y *= NEG_INV_TAU;
        if (c2 < 128) *(float2*)(out_y1 + (size_t)(rbase + r) * 128 + c2) = v;
        else          *(float2*)(out_y2 + (size_t)(rbase + r) * 64 + (c2 - 128)) = v;
    }
}

extern "C" void kernel_launch(void* const* d_in, const int* in_sizes, int n_in,
                              void* d_out, int out_size, void* d_ws, size_t ws_size,
                              hipStream_t stream) {
    const float* x0 = (const float*)d_in[0];
    const float* y1 = (const float*)d_in[1];
    const float* y2 = (const float*)d_in[2];
    const float* z  = (const float*)d_in[3];
    const float* c  = (const float*)d_in[4];
    const float* A1 = (const float*)d_in[5];
    const float* A2 = (const float*)d_in[6];
    (void)in_sizes; (void)n_in; (void)out_size; (void)d_ws; (void)ws_size;
    pdhg_kernel<<<dim3(8192 / 32), dim3(128), 0, stream>>>(
        x0, y1, y2, z, c, A1, A2, (float*)d_out);
}